// SphericalConv2D_11330123727304
// MI455X (gfx1250) — compile-verified
//
#include <hip/hip_runtime.h>
#include <hip/hip_bf16.h>
#include <math.h>

typedef __attribute__((ext_vector_type(2))) float v2f;
typedef __attribute__((ext_vector_type(8))) float v8f;

constexpr int BATCH = 2;
constexpr int CIN   = 8;
constexpr int H     = 512;
constexpr int W     = 1024;
constexpr int OUTC  = 8;
constexpr int KK    = 9;             // 3x3 taps
constexpr int HW    = H * W;
constexpr int CK    = CIN * KK;      // 72 = contraction length
constexpr float PI_F = 3.14159265358979323846f;

// 4-tap bilinear blend from one channel plane; offsets precomputed (include lane-half channel offset)
__device__ __forceinline__ float blend4(const float* __restrict__ pl,
                                        int o00, int o01, int o10, int o11,
                                        float w00, float w01, float w10, float w11) {
  return pl[o00] * w00 + pl[o01] * w01 + pl[o10] * w10 + pl[o11] * w11;
}

// xA = x + 2*HW               (batch 0 base, centered at plane 2)
// xB = x + CIN*HW + 2*HW      (batch 1 base, centered at plane 2)
// Passing them as separate kernargs keeps them in distinct SGPR pairs so every
// gather lowers to saddr + voffset*4 + imm24 with imm in {-4MB,-2MB,+4MB,+6MB}.
__global__ void __launch_bounds__(256)
spherical_conv2d_wmma(const float* __restrict__ xA,
                      const float* __restrict__ xB,
                      const float* __restrict__ weight,
                      const float* __restrict__ bias,
                      float* __restrict__ out) {
  const int lane = threadIdx.x & 31;
  const int wave = threadIdx.x >> 5;
  const int pixBase = (blockIdx.x * 8 + wave) * 16;   // 16-pixel tile per wave

  const int m  = lane & 15;     // A-row (pixel in tile) / B-col (out channel)
  const int hi = lane >> 4;     // lane half: selects channel pair within chunk
  const int p  = pixBase + m;
  const int row = p >> 10;      // / W
  const int col = p & (W - 1);

  // ---- per-pixel tangent-plane axes (once per lane, reused for all taps/channels/batches) ----
  float az = (float)(2 * col + 1 - W) * (PI_F / (float)W);
  float po = (float)(2 * row + 1) * (PI_F / (float)(2 * H));
  float sa = __sinf(az), ca = __cosf(az);
  float sp = __sinf(po), cp = __cosf(po);
  const float zx = sp * sa, zy = -cp, zz = sp * ca;   // axis_z
  const float axx = ca,              axz = -sa;       // axis_x (y comp = 0)
  const float ayx = cp * sa, ayy = sp, ayz = cp * ca; // axis_y
  constexpr float kscale = PI_F / (float)H;

  const int chOff = 2 * hi * HW;     // this lane-half's channel-pair element offset (in voffset)
  const int o     = m;               // output channel (B-fragment column)
  const float bv  = bias[o & (OUTC - 1)];
  // Lanes with o>=8 read duplicate (in-bounds) weights; they only feed D columns
  // that are never stored, so no zero-masking / predication is needed.
  const float* wt = weight + (o & (OUTC - 1)) * CK;   // weight[o][c*9+k]

  v8f acc0 = {0.0f, 0.0f, 0.0f, 0.0f, 0.0f, 0.0f, 0.0f, 0.0f};
  v8f acc1 = {0.0f, 0.0f, 0.0f, 0.0f, 0.0f, 0.0f, 0.0f, 0.0f};

  // Contraction permuted to K' = k*8 + c:
  //   chunk 2t   covers (tap t, channels 0..3):  lane half hi -> c = 2*hi, 2*hi+1
  //   chunk 2t+1 covers (tap t, channels 4..7):  lane half hi -> c = 4+2*hi, 4+2*hi+1
  // Same permutation applied to A and B keeps D = A*B^T exact.
#pragma unroll
  for (int t = 0; t < KK; ++t) {
    // ---- bilinear descriptor for tap t (shared by 8 channels x 2 batches) ----
    float dx = (float)(t % 3 - 1) * kscale;
    float dy = (float)(t / 3 - 1) * kscale;
    float X = zx + axx * dx + ayx * dy;
    float Y = zy + ayy * dy;
    float Z = zz + axz * dx + ayz * dy;
    float pol = atan2f(sqrtf(X * X + Z * Z), -Y);
    float azs = atan2f(X, Z);
    float gx = (azs + PI_F) * ((float)W / (2.0f * PI_F)) - 0.5f;
    float gy = pol * ((float)H / PI_F) - 0.5f;
    float x0f = floorf(gx), y0f = floorf(gy);
    float fxw = gx - x0f, fyw = gy - y0f;
    int x0 = ((int)x0f) & (W - 1);
    int x1 = (x0 + 1) & (W - 1);
    int y0 = (int)y0f;
    int y0c = min(max(y0, 0), H - 1);
    int y1c = min(max(y0 + 1, 0), H - 1);
    int o00 = y0c * W + x0 + chOff, o01 = y0c * W + x1 + chOff;
    int o10 = y1c * W + x0 + chOff, o11 = y1c * W + x1 + chOff;
    float w00 = (1.0f - fxw) * (1.0f - fyw), w01 = fxw * (1.0f - fyw);
    float w10 = (1.0f - fxw) * fyw,          w11 = fxw * fyw;

    // ---- B fragments for this tap (permuted indexing, unconditional loads) ----
    v2f BA, BB;
    BA.x = wt[(2 * hi + 0) * KK + t];
    BA.y = wt[(2 * hi + 1) * KK + t];
    BB.x = wt[(2 * hi + 4) * KK + t];
    BB.y = wt[(2 * hi + 5) * KK + t];

    // Plane-group constants relative to the centered batch bases:
    //   g=0 -> -2*HW, g=1 -> -1*HW, g=4 -> +2*HW, g=5 -> +3*HW   (all fit imm24)
    // ---- chunk 2t (channels 0..3), both batches, interleaved acc chains ----
    v2f A;
    A.x = blend4(xA - 2 * HW, o00, o01, o10, o11, w00, w01, w10, w11);
    A.y = blend4(xA - 1 * HW, o00, o01, o10, o11, w00, w01, w10, w11);
    acc0 = __builtin_amdgcn_wmma_f32_16x16x4_f32(false, A, false, BA, (short)0, acc0, false, false);

    A.x = blend4(xB - 2 * HW, o00, o01, o10, o11, w00, w01, w10, w11);
    A.y = blend4(xB - 1 * HW, o00, o01, o10, o11, w00, w01, w10, w11);
    acc1 = __builtin_amdgcn_wmma_f32_16x16x4_f32(false, A, false, BA, (short)0, acc1, false, false);

    // ---- chunk 2t+1 (channels 4..7) ----
    A.x = blend4(xA + 2 * HW, o00, o01, o10, o11, w00, w01, w10, w11);
    A.y = blend4(xA + 3 * HW, o00, o01, o10, o11, w00, w01, w10, w11);
    acc0 = __builtin_amdgcn_wmma_f32_16x16x4_f32(false, A, false, BB, (short)0, acc0, false, false);

    A.x = blend4(xB + 2 * HW, o00, o01, o10, o11, w00, w01, w10, w11);
    A.y = blend4(xB + 3 * HW, o00, o01, o10, o11, w00, w01, w10, w11);
    acc1 = __builtin_amdgcn_wmma_f32_16x16x4_f32(false, A, false, BB, (short)0, acc1, false, false);
  }

  // D layout: acc[r] -> pixel pixBase + r + 8*hi, channel o = lane&15
  if (o < OUTC) {
    float* ob0 = out + (size_t)o * HW          + pixBase + hi * 8;
    float* ob1 = out + (size_t)(OUTC + o) * HW + pixBase + hi * 8;
#pragma unroll
    for (int r = 0; r < 8; ++r) {
      ob0[r] = acc0[r] + bv;
      ob1[r] = acc1[r] + bv;
    }
  }
}

extern "C" void kernel_launch(void* const* d_in, const int* in_sizes, int n_in,
                              void* d_out, int out_size, void* d_ws, size_t ws_size,
                              hipStream_t stream) {
  const float* x      = (const float*)d_in[0];
  const float* weight = (const float*)d_in[1];
  const float* bias   = (const float*)d_in[2];
  float* out = (float*)d_out;
  // Centered batch base pointers (see kernel comment): keep them as independent
  // kernargs so the backend cannot re-fold them into one out-of-range base.
  const float* xA = x + 2 * HW;
  const float* xB = x + (size_t)CIN * HW + 2 * HW;
  // HW/16 = 32768 tiles, 8 waves (tiles) per 256-thread block -> 4096 blocks
  spherical_conv2d_wmma<<<(HW / 16) / 8, 256, 0, stream>>>(xA, xB, weight, bias, out);
}